// Involution_68255620268915
// MI455X (gfx1250) — compile-verified
//
#include <hip/hip_runtime.h>
#include <math.h>

typedef __attribute__((ext_vector_type(2))) float v2f;
typedef __attribute__((ext_vector_type(8))) float v8f;

// Problem constants (from reference)
constexpr int kK   = 7;        // kernel size
constexpr int kG   = 16;       // groups
constexpr int kP   = 3;        // padding
constexpr int kB   = 8;
constexpr int kC   = 256;
constexpr int kCR  = 64;       // reduced channels
constexpr int kKKG = kK * kK * kG;  // 784 = 49 tiles of 16
constexpr int kH   = 64;
constexpr int kW   = 64;
constexpr int kHW  = kH * kW;  // 4096
constexpr float kEPS = 1e-5f;

// Workspace layout (in floats)
constexpr size_t WS_WR = 0;                       // folded reduce weights: 64*256
constexpr size_t WS_BR = WS_WR + (size_t)kCR*kC;  // folded reduce bias: 64
constexpr size_t WS_WS = WS_BR + kCR;             // normed span weights: 784*64
constexpr size_t WS_R  = WS_WS + (size_t)kKKG*kCR;// r activations: 8*64*4096

// ---------------------------------------------------------------------------
// Kernel 1: weight-norm both convs; fold BN (eval mode) into reduce weights.
//   w_r'[o][c] = v_r[o][c] * g_r[o]/||v_r[o]|| * gamma[o]/sqrt(var[o]+eps)
//   b_r'[o]    = (b_r[o]-mean[o])*gamma[o]/sqrt(var[o]+eps) + beta[o]
//   w_s'[o][c] = v_s[o][c] * g_s[o]/||v_s[o]||
// One block per output row; LDS tree reduction for the row norm.
// ---------------------------------------------------------------------------
__global__ void inv_prep_weights(const float* __restrict__ v_reduce,
                                 const float* __restrict__ g_reduce,
                                 const float* __restrict__ b_reduce,
                                 const float* __restrict__ bn_gamma,
                                 const float* __restrict__ bn_beta,
                                 const float* __restrict__ bn_mean,
                                 const float* __restrict__ bn_var,
                                 const float* __restrict__ v_span,
                                 const float* __restrict__ g_span,
                                 float* __restrict__ ws) {
    __shared__ float red[256];
    const int tid = threadIdx.x;
    const int row = blockIdx.x;
    if (row < kCR) {
        float v = v_reduce[row * kC + tid];
        red[tid] = v * v;
        __syncthreads();
        for (int s = 128; s > 0; s >>= 1) {
            if (tid < s) red[tid] += red[tid + s];
            __syncthreads();
        }
        const float norm   = sqrtf(red[0]);
        const float bscale = bn_gamma[row] * rsqrtf(bn_var[row] + kEPS);
        ws[WS_WR + (size_t)row * kC + tid] = v * (g_reduce[row] / norm) * bscale;
        if (tid == 0)
            ws[WS_BR + row] = (b_reduce[row] - bn_mean[row]) * bscale + bn_beta[row];
    } else {
        const int o = row - kCR;
        float v = (tid < kCR) ? v_span[o * kCR + tid] : 0.0f;
        red[tid] = v * v;
        __syncthreads();
        for (int s = 128; s > 0; s >>= 1) {
            if (tid < s) red[tid] += red[tid + s];
            __syncthreads();
        }
        const float norm = sqrtf(red[0]);
        if (tid < kCR)
            ws[WS_WS + (size_t)o * kCR + tid] = v * (g_span[o] / norm);
    }
}

// ---------------------------------------------------------------------------
// Kernel 2: reduce GEMM with fp32 WMMA.
//   r[b][m][n] = relu( sum_k WR[m][k] * x[b][k][n] + BR[m] )
//   M=64 (CR), K=256 (C), N=4096 (HW) per batch image.
// Block = 256 threads = 8 waves; wave (mt,nt) does a 16x16 D tile.
// WMMA 16x16x4 per-lane layout (wave32):
//   A: lane l -> row l&15, K-pair at (l>>4)*2  (2 VGPRs)
//   B: lane l -> col l&15, K-pair at (l>>4)*2  (2 VGPRs)
//   D: acc[i] -> (M = i + 8*(l>>4), N = l&15)
// ---------------------------------------------------------------------------
__global__ void inv_reduce_gemm(const float* __restrict__ x,
                                const float* __restrict__ ws,
                                float* __restrict__ r) {
    const float* WR = ws + WS_WR;
    const float* BR = ws + WS_BR;
    const int lane = threadIdx.x & 31;
    const int wave = threadIdx.x >> 5;
    const int mt = wave & 3;          // 4 M tiles -> 64 rows
    const int nt = wave >> 2;         // 2 N tiles -> 32 pixels/block
    const int b  = blockIdx.y;
    const int n0 = blockIdx.x * 32 + nt * 16;
    const int m0 = mt * 16;
    const int rc = lane & 15;         // row (A) / col (B)
    const int kh = (lane >> 4) * 2;   // K sub-offset per half-wave
    const float* xb = x + (size_t)b * kC * kHW;

    v8f acc = {};
    for (int k0 = 0; k0 < kC; k0 += 4) {
        v2f a, bb;
        a.x  = WR[(m0 + rc) * kC + k0 + kh];
        a.y  = WR[(m0 + rc) * kC + k0 + kh + 1];
        bb.x = xb[(size_t)(k0 + kh) * kHW + n0 + rc];
        bb.y = xb[(size_t)(k0 + kh + 1) * kHW + n0 + rc];
        acc = __builtin_amdgcn_wmma_f32_16x16x4_f32(false, a, false, bb,
                                                    (short)0, acc, false, false);
    }

    float* rb = r + (size_t)b * kCR * kHW;
    const int nn  = n0 + (lane & 15);
    const int mhi = 8 * (lane >> 4);
    for (int i = 0; i < 8; ++i) {
        const int m = m0 + i + mhi;
        float v = acc[i] + BR[m];
        rb[(size_t)m * kHW + nn] = v > 0.f ? v : 0.f;
    }
}

// ---------------------------------------------------------------------------
// Kernel 3: fused span GEMM (WMMA) + involution.
// One block per (batch, image-row h): 64-pixel tile.
// Phase 1: ker[o][p] for o in [0,784), p in [0,64) via WMMA into LDS (196 KB).
//          784 rows = 49 M-tiles of 16; M-tile == kernel tap (ki,kj),
//          row within tile == group g  (o = (ki*7+kj)*16 + g).
// Phase 2: out[b][c][h][w] = sum_{ki,kj} x[b][c][h+ki-3][w+kj-3] *
//                            ker[(ki*7+kj)*16 + c/16][w]   (x reads hit L2).
// ---------------------------------------------------------------------------
__global__ void inv_span_involution(const float* __restrict__ x,
                                    const float* __restrict__ ws,
                                    const float* __restrict__ b_span,
                                    float* __restrict__ out) {
    extern __shared__ float lds[];    // [784][64] per-pixel kernels
    const float* WSP = ws + WS_WS;
    const float* R   = ws + WS_R;
    const int tid  = threadIdx.x;
    const int lane = tid & 31;
    const int wave = tid >> 5;
    const int b    = blockIdx.y;
    const int h    = blockIdx.x;      // tile = full image row h (64 pixels)
    const int hw0  = h * kW;
    const float* Rb = R + (size_t)b * kCR * kHW;

    const int rc  = lane & 15;
    const int kh  = (lane >> 4) * 2;
    const int mhi = 8 * (lane >> 4);

    // Phase 1: 49 M-tiles x 4 N-tiles = 196 WMMA tiles across 8 waves.
    for (int t = wave; t < 49 * 4; t += 8) {
        const int mt   = t >> 2;      // tap index kk = ki*7+kj
        const int nt   = t & 3;
        const int m0   = mt * 16;
        const int nloc = nt * 16;
        v8f acc = {};
        for (int k0 = 0; k0 < kCR; k0 += 4) {
            v2f a, bb;
            a.x  = WSP[(m0 + rc) * kCR + k0 + kh];
            a.y  = WSP[(m0 + rc) * kCR + k0 + kh + 1];
            bb.x = Rb[(size_t)(k0 + kh) * kHW + hw0 + nloc + rc];
            bb.y = Rb[(size_t)(k0 + kh + 1) * kHW + hw0 + nloc + rc];
            acc = __builtin_amdgcn_wmma_f32_16x16x4_f32(false, a, false, bb,
                                                        (short)0, acc, false, false);
        }
        const int col = lane & 15;
        for (int i = 0; i < 8; ++i) {
            const int m = m0 + i + mhi;              // o = kk*16 + g
            lds[m * 64 + nloc + col] = acc[i] + b_span[m];
        }
    }
    __syncthreads();

    // Phase 2: 256 channels x 64 pixels = 16384 outputs, 64 per thread.
    const float* xb = x + (size_t)b * kC * kHW;
    float* ob = out + (size_t)b * kC * kHW;
    for (int it = 0; it < 64; ++it) {
        const int e = it * 256 + tid;
        const int c = e >> 6;         // channel
        const int p = e & 63;         // w coordinate within row
        const int g = c >> 4;         // group = c / (C/G)
        const float* xc = xb + (size_t)c * kHW;
        const float* kg = lds + g * 64 + p;
        float acc = 0.f;
        for (int ki = 0; ki < kK; ++ki) {
            const int hh = h + ki - kP;
            if (hh < 0 || hh >= kH) continue;
            const float* xrow = xc + hh * kW;
            for (int kj = 0; kj < kK; ++kj) {
                const int wwj = p + kj - kP;
                if (wwj < 0 || wwj >= kW) continue;
                acc += xrow[wwj] * kg[(ki * kK + kj) * (kG * 64)];
            }
        }
        ob[(size_t)c * kHW + hw0 + p] = acc;
    }
}

extern "C" void kernel_launch(void* const* d_in, const int* in_sizes, int n_in,
                              void* d_out, int out_size, void* d_ws, size_t ws_size,
                              hipStream_t stream) {
    (void)in_sizes; (void)n_in; (void)out_size; (void)ws_size;
    const float* x        = (const float*)d_in[0];
    const float* v_reduce = (const float*)d_in[1];
    const float* g_reduce = (const float*)d_in[2];
    const float* b_reduce = (const float*)d_in[3];
    const float* bn_gamma = (const float*)d_in[4];
    const float* bn_beta  = (const float*)d_in[5];
    const float* bn_mean  = (const float*)d_in[6];
    const float* bn_var   = (const float*)d_in[7];
    const float* v_span   = (const float*)d_in[8];
    const float* g_span   = (const float*)d_in[9];
    const float* b_span   = (const float*)d_in[10];
    float* out = (float*)d_out;
    float* ws  = (float*)d_ws;

    // 1) weight-norm + BN folding (tiny)
    inv_prep_weights<<<kCR + kKKG, 256, 0, stream>>>(
        v_reduce, g_reduce, b_reduce, bn_gamma, bn_beta, bn_mean, bn_var,
        v_span, g_span, ws);

    // 2) reduce GEMM -> r  (M=64, K=256, N=32768), WMMA f32 16x16x4
    inv_reduce_gemm<<<dim3(kHW / 32, kB), 256, 0, stream>>>(x, ws, ws + WS_R);

    // 3) fused span GEMM + involution; 196 KB dynamic LDS for per-pixel kernels
    inv_span_involution<<<dim3(kH, kB), 256, kKKG * 64 * sizeof(float), stream>>>(
        x, ws, b_span, out);
}